// Generator_9809705304253
// MI455X (gfx1250) — compile-verified
//
#include <hip/hip_runtime.h>
#include <hip/hip_bf16.h>

#define B_     2
#define LSEQ   1024
#define EMB    384
#define DM     768
#define DI     1536
#define NST    24
#define DTR    48
#define DCONV  4
#define NLAYERS 4
#define XD     (DTR + 2*NST)   /* 96 */
#define NTOK   (B_*LSEQ)       /* 2048 */
#define LABEL  1001

typedef __attribute__((ext_vector_type(2))) float v2f;
typedef __attribute__((ext_vector_type(8))) float v8f;
typedef __attribute__((ext_vector_type(4))) int   v4i;

__device__ __forceinline__ float silu_f(float x) { return x / (1.f + __expf(-x)); }

// ---- CDNA5 async global->LDS copy (ASYNCcnt path), with safe sync fallback
#if defined(__gfx1250__) && __has_builtin(__builtin_amdgcn_global_load_async_to_lds_b128)
#define HAVE_ASYNC 1
typedef __attribute__((address_space(1))) v4i gv4i;   // global v4i
typedef __attribute__((address_space(3))) v4i lv4i;   // LDS v4i
#define ASYNC_COPY_B128(dst, src)                                              \
  __builtin_amdgcn_global_load_async_to_lds_b128((gv4i*)(src), (lv4i*)(dst), 0, 0)
#if __has_builtin(__builtin_amdgcn_s_wait_asynccnt)
#define ASYNC_WAIT() __builtin_amdgcn_s_wait_asynccnt(0)
#else
#define ASYNC_WAIT() asm volatile("s_wait_asynccnt 0x0" ::: "memory")
#endif
#else
#define HAVE_ASYNC 0
#define ASYNC_WAIT()
#endif

// ------------------------------------------------------------------ embedding
__global__ __launch_bounds__(256) void embed_kernel(
    const int* __restrict__ inputs, const int* __restrict__ relation,
    const float* __restrict__ table, float* __restrict__ x)
{
  int i = blockIdx.x * 256 + threadIdx.x;      // over NTOK*DM
  if (i >= NTOK * DM) return;
  int d = i % DM;
  int t = i / DM;
  int b = t / LSEQ;
  int tok = (d < EMB) ? inputs[t] : relation[b];
  int dd  = (d < EMB) ? d : d - EMB;
  x[i] = table[(size_t)tok * EMB + dd];
}

// ------------------------------------------------------------------ layernorm
// one wave (32 lanes) per row of DM=768
__global__ __launch_bounds__(256) void ln_kernel(
    const float* __restrict__ x, const float* __restrict__ w,
    const float* __restrict__ b, float* __restrict__ out)
{
  int row  = (blockIdx.x * 256 + threadIdx.x) >> 5;
  int lane = threadIdx.x & 31;
  if (row >= NTOK) return;
  const float* xr = x + (size_t)row * DM;
  float s = 0.f;
  #pragma unroll
  for (int i = lane; i < DM; i += 32) s += xr[i];
  #pragma unroll
  for (int off = 16; off > 0; off >>= 1) s += __shfl_xor(s, off, 32);
  float mean = s * (1.f / DM);
  float v = 0.f;
  #pragma unroll
  for (int i = lane; i < DM; i += 32) { float t = xr[i] - mean; v += t * t; }
  #pragma unroll
  for (int off = 16; off > 0; off >>= 1) v += __shfl_xor(v, off, 32);
  float rstd = rsqrtf(v * (1.f / DM) + 1e-5f);
  for (int i = lane; i < DM; i += 32)
    out[(size_t)row * DM + i] = (xr[i] - mean) * rstd * w[i] + b[i];
}

// ------------------------------------------------------------------ WMMA GEMM
// C[M,N] = A[M,K] @ W^T  (W is [N,K] row-major), M = NTOK (multiple of 128)
// mode: 0 = store, 1 = +bias, 2 = softplus(gemm + bias), 3 = resid + gemm
// 128x64 block, 8 waves, each wave owns a 32x32 patch (2x2 WMMA tiles).
// Double-buffered LDS stages filled by async global->LDS B128 copies.
#define BM 128
#define BN 64
#define BK 16
#define LDP (BK + 4)   /* pitch 20 floats: 16B-aligned rows, conflict-free */

__global__ __launch_bounds__(256) void gemm_wmma_f32(
    const float* __restrict__ A, int lda,
    const float* __restrict__ W,
    float* __restrict__ C, int ldc,
    int N, int K,
    const float* __restrict__ bias,
    const float* __restrict__ resid,
    int mode)
{
  __shared__ float As[2][BM][LDP];
  __shared__ float Bs[2][BN][LDP];

  const int tid    = threadIdx.x;
  const int lane   = tid & 31;
  const int wave   = tid >> 5;
  const int laneLo = lane & 15;
  const int laneHi = lane >> 4;
  const int m0 = blockIdx.y * BM;
  const int n0 = blockIdx.x * BN;
  const int mw = (wave & 3) * 2;     // m-tiles mw, mw+1   (of 8)
  const int nw = (wave >> 2) * 2;    // n-tiles nw, nw+1   (of 4)

  v8f c00 = {0.f,0.f,0.f,0.f,0.f,0.f,0.f,0.f};
  v8f c01 = c00, c10 = c00, c11 = c00;

  // stage-load indexing: A = 512 float4 (2/thread), B = 256 float4 (1/thread)
  const int arow0 = tid >> 2;        // 0..63
  const int arow1 = arow0 + 64;      // 64..127
  const int kq    = (tid & 3) * 4;   // 0,4,8,12
  int gn = n0 + arow0;               // B row; clamp OOB (cols guarded at store)
  if (gn >= N) gn = N - 1;

  auto issue = [&](int buf, int k0) {
    const float* a0p = A + (size_t)(m0 + arow0) * lda + k0 + kq;
    const float* a1p = A + (size_t)(m0 + arow1) * lda + k0 + kq;
    const float* bp  = W + (size_t)gn * K + k0 + kq;
#if HAVE_ASYNC
    ASYNC_COPY_B128(&As[buf][arow0][kq], a0p);
    ASYNC_COPY_B128(&As[buf][arow1][kq], a1p);
    ASYNC_COPY_B128(&Bs[buf][arow0][kq], bp);
#else
    float4 va0 = *reinterpret_cast<const float4*>(a0p);
    float4 va1 = *reinterpret_cast<const float4*>(a1p);
    float4 vb  = *reinterpret_cast<const float4*>(bp);
    *reinterpret_cast<float4*>(&As[buf][arow0][kq]) = va0;
    *reinterpret_cast<float4*>(&As[buf][arow1][kq]) = va1;
    *reinterpret_cast<float4*>(&Bs[buf][arow0][kq]) = vb;
#endif
  };

  issue(0, 0);
  int buf = 0;
  for (int k0 = 0; k0 < K; k0 += BK) {
    ASYNC_WAIT();        // our async LDS writes for `buf` have landed
    __syncthreads();     // everyone's writes visible; prev reads of buf^1 done
    if (k0 + BK < K) issue(buf ^ 1, k0 + BK);

    #pragma unroll
    for (int kk = 0; kk < 4; ++kk) {
      const int kb = kk * 4 + laneHi * 2;
      v2f a0, a1, b0, b1;
      a0.x = As[buf][(mw + 0) * 16 + laneLo][kb];
      a0.y = As[buf][(mw + 0) * 16 + laneLo][kb + 1];
      a1.x = As[buf][(mw + 1) * 16 + laneLo][kb];
      a1.y = As[buf][(mw + 1) * 16 + laneLo][kb + 1];
      b0.x = Bs[buf][(nw + 0) * 16 + laneLo][kb];
      b0.y = Bs[buf][(nw + 0) * 16 + laneLo][kb + 1];
      b1.x = Bs[buf][(nw + 1) * 16 + laneLo][kb];
      b1.y = Bs[buf][(nw + 1) * 16 + laneLo][kb + 1];
      c00 = __builtin_amdgcn_wmma_f32_16x16x4_f32(false, a0, false, b0, (short)0, c00, false, false);
      c01 = __builtin_amdgcn_wmma_f32_16x16x4_f32(false, a0, false, b1, (short)0, c01, false, false);
      c10 = __builtin_amdgcn_wmma_f32_16x16x4_f32(false, a1, false, b0, (short)0, c10, false, false);
      c11 = __builtin_amdgcn_wmma_f32_16x16x4_f32(false, a1, false, b1, (short)0, c11, false, false);
    }
    buf ^= 1;
  }

  // epilogue: VGPR r -> (M = r + 8*laneHi, N = laneLo) within each 16x16 tile
  auto epi = [&](const v8f& c, int mt, int nt) {
    int col = n0 + nt * 16 + laneLo;
    if (col >= N) return;
    float bcol = (mode == 1 || mode == 2) ? bias[col] : 0.f;
    #pragma unroll
    for (int r = 0; r < 8; ++r) {
      int row = m0 + mt * 16 + r + laneHi * 8;
      float v = c[r] + bcol;
      if (mode == 2) v = (v > 20.f) ? v : log1pf(__expf(v));
      if (mode == 3) v += resid[(size_t)row * ldc + col];
      C[(size_t)row * ldc + col] = v;
    }
  };
  epi(c00, mw, nw);     epi(c01, mw, nw + 1);
  epi(c10, mw + 1, nw); epi(c11, mw + 1, nw + 1);
}

// ------------------------------------------------------------ depthwise conv
__global__ __launch_bounds__(256) void conv_silu_kernel(
    const float* __restrict__ xz, const float* __restrict__ cw,
    const float* __restrict__ cb, float* __restrict__ xc)
{
  int i = blockIdx.x * 256 + threadIdx.x;     // over NTOK*DI
  if (i >= NTOK * DI) return;
  int d = i % DI;
  int t = i / DI;
  int l = t % LSEQ;
  int b = t / LSEQ;
  float acc = cb[d];
  #pragma unroll
  for (int k = 0; k < DCONV; ++k) {
    int ll = l - (DCONV - 1) + k;
    if (ll >= 0)
      acc += xz[((size_t)b * LSEQ + ll) * (2 * DI) + d] * cw[d * DCONV + k];
  }
  xc[(size_t)t * DI + d] = silu_f(acc);
}

// ------------------------------------------------------------ selective scan
// 8 lanes cooperate per channel d: lane slot nn holds states {nn, nn+8, nn+16};
// per-step reduction over the 8 slots via 3 shfl_xor hops. 4 channels / wave.
__global__ __launch_bounds__(256) void scan_kernel(
    const float* __restrict__ xc, const float* __restrict__ dlt,
    const float* __restrict__ xdbl, const float* __restrict__ xz,
    const float* __restrict__ A_log, const float* __restrict__ Dp,
    float* __restrict__ yg)
{
  const int lane = threadIdx.x & 31;
  const int wv   = threadIdx.x >> 5;
  const int nn   = lane & 7;          // state slot
  const int sub  = lane >> 3;         // channel within wave (0..3)
  const int d = blockIdx.x * 32 + wv * 4 + sub;   // 0..DI-1
  const int b = blockIdx.y;

  float Ac0 = -__expf(A_log[(size_t)d * NST + nn]);
  float Ac1 = -__expf(A_log[(size_t)d * NST + nn + 8]);
  float Ac2 = -__expf(A_log[(size_t)d * NST + nn + 16]);
  float h0 = 0.f, h1 = 0.f, h2 = 0.f;
  float Dd = Dp[d];

  const float* dptr = dlt  + (size_t)b * LSEQ * DI + d;
  const float* uptr = xc   + (size_t)b * LSEQ * DI + d;
  const float* zptr = xz   + (size_t)b * LSEQ * (2 * DI) + DI + d;
  const float* bc   = xdbl + (size_t)b * LSEQ * XD;
  float*       yptr = yg   + (size_t)b * LSEQ * DI + d;

  for (int l = 0; l < LSEQ; ++l) {
    float dt = dptr[(size_t)l * DI];
    float u  = uptr[(size_t)l * DI];
    float du = dt * u;
    const float* Bl = bc + (size_t)l * XD + DTR;
    const float* Cl = Bl + NST;
    h0 = __expf(dt * Ac0) * h0 + du * Bl[nn];
    h1 = __expf(dt * Ac1) * h1 + du * Bl[nn + 8];
    h2 = __expf(dt * Ac2) * h2 + du * Bl[nn + 16];
    float y = h0 * Cl[nn] + h1 * Cl[nn + 8] + h2 * Cl[nn + 16];
    y += __shfl_xor(y, 1, 32);
    y += __shfl_xor(y, 2, 32);
    y += __shfl_xor(y, 4, 32);
    if (nn == 0) {
      y += u * Dd;
      float z = zptr[(size_t)l * (2 * DI)];
      yptr[(size_t)l * DI] = y * silu_f(z);
    }
  }
}

// ------------------------------------------------------------------ launcher
extern "C" void kernel_launch(void* const* d_in, const int* in_sizes, int n_in,
                              void* d_out, int out_size, void* d_ws, size_t ws_size,
                              hipStream_t stream) {
  const int*   inputs     = (const int*)  d_in[0];
  const int*   relation   = (const int*)  d_in[1];
  const float* emb_table  = (const float*)d_in[2];
  const float* ln_w       = (const float*)d_in[3];
  const float* ln_b       = (const float*)d_in[4];
  const float* in_proj_w  = (const float*)d_in[5];
  const float* conv_w     = (const float*)d_in[6];
  const float* conv_b     = (const float*)d_in[7];
  const float* x_proj_w   = (const float*)d_in[8];
  const float* dt_proj_w  = (const float*)d_in[9];
  const float* dt_proj_b  = (const float*)d_in[10];
  const float* A_log      = (const float*)d_in[11];
  const float* D_param    = (const float*)d_in[12];
  const float* out_proj_w = (const float*)d_in[13];
  const float* fn_w       = (const float*)d_in[14];
  const float* fn_b       = (const float*)d_in[15];
  const float* head_w     = (const float*)d_in[16];
  const float* head_b     = (const float*)d_in[17];
  float* out = (float*)d_out;

  float* ws = (float*)d_ws;
  size_t off = 0;
  float* x    = ws + off; off += (size_t)NTOK * DM;       // residual stream
  float* hbuf = ws + off; off += (size_t)NTOK * DM;       // LN output
  float* xz   = ws + off; off += (size_t)NTOK * 2 * DI;   // in_proj out
  float* xc   = ws + off; off += (size_t)NTOK * DI;       // conv+silu out
  float* xdbl = ws + off; off += (size_t)NTOK * XD;       // x_proj out
  float* dlt  = ws + off; off += (size_t)NTOK * DI;       // softplus(dt)
  float* ygb  = ws + off; off += (size_t)NTOK * DI;       // gated scan out

  embed_kernel<<<(NTOK * DM + 255) / 256, 256, 0, stream>>>(inputs, relation, emb_table, x);

  for (int i = 0; i < NLAYERS; ++i) {
    ln_kernel<<<NTOK / 8, 256, 0, stream>>>(x, ln_w + i * DM, ln_b + i * DM, hbuf);

    dim3 g1((2 * DI + BN - 1) / BN, NTOK / BM);
    gemm_wmma_f32<<<g1, 256, 0, stream>>>(hbuf, DM, in_proj_w + (size_t)i * 2 * DI * DM,
                                          xz, 2 * DI, 2 * DI, DM, nullptr, nullptr, 0);

    conv_silu_kernel<<<(NTOK * DI + 255) / 256, 256, 0, stream>>>(
        xz, conv_w + (size_t)i * DI * DCONV, conv_b + (size_t)i * DI, xc);

    dim3 g2((XD + BN - 1) / BN, NTOK / BM);
    gemm_wmma_f32<<<g2, 256, 0, stream>>>(xc, DI, x_proj_w + (size_t)i * XD * DI,
                                          xdbl, XD, XD, DI, nullptr, nullptr, 0);

    dim3 g3((DI + BN - 1) / BN, NTOK / BM);
    gemm_wmma_f32<<<g3, 256, 0, stream>>>(xdbl, XD, dt_proj_w + (size_t)i * DI * DTR,
                                          dlt, DI, DI, DTR, dt_proj_b + (size_t)i * DI,
                                          nullptr, 2);

    dim3 gs(DI / 32, B_);
    scan_kernel<<<gs, 256, 0, stream>>>(xc, dlt, xdbl, xz,
                                        A_log + (size_t)i * DI * NST,
                                        D_param + (size_t)i * DI, ygb);

    dim3 g4((DM + BN - 1) / BN, NTOK / BM);
    gemm_wmma_f32<<<g4, 256, 0, stream>>>(ygb, DI, out_proj_w + (size_t)i * DM * DI,
                                          x, DM, DM, DI, nullptr, x, 3);
  }

  ln_kernel<<<NTOK / 8, 256, 0, stream>>>(x, fn_w, fn_b, hbuf);
  dim3 gh((LABEL + BN - 1) / BN, NTOK / BM);
  gemm_wmma_f32<<<gh, 256, 0, stream>>>(hbuf, DM, head_w, out, LABEL,
                                        LABEL, DM, head_b, nullptr, 1);
}